// GNNChannel_30812095382125
// MI455X (gfx1250) — compile-verified
//
#include <hip/hip_runtime.h>
#include <hip/hip_bf16.h>

typedef __attribute__((ext_vector_type(2))) float v2f;
typedef __attribute__((ext_vector_type(8))) float v8f;

#define D 128
#define NEG_SLOPE 0.2f

// ---------------------------------------------------------------------------
// Init: acc = 0, mx = -inf, denom = 0  (re-done every call: ws is not
// re-poisoned between replays, so we must establish our own state).
// ---------------------------------------------------------------------------
__global__ void __launch_bounds__(256) k_init(float* __restrict__ acc,
                                              float* __restrict__ mx,
                                              float* __restrict__ denom,
                                              int nNodes, int total) {
  int i = blockIdx.x * blockDim.x + threadIdx.x;
  if (i < total) acc[i] = 0.0f;
  if (i < nNodes) {
    mx[i] = -__builtin_inff();
    denom[i] = 0.0f;
  }
}

// ---------------------------------------------------------------------------
// Dual GEMM with fp32 WMMA:  xl = h @ Wl,  xr = h @ Wr.
// One wave (32 lanes) per 16-row slab; wave computes all 8 column tiles for
// both weight matrices, reusing the A fragment.  V_WMMA_F32_16X16X4_F32,
// K-loop of 32 steps (K=4 each).
// A frag layout (ISA 7.12.2, 32-bit A 16x4): lanes 0-15 hold K=k,k+1 for
// M=lane; lanes 16-31 hold K=k+2,k+3 for M=lane-16.  B mirrored.  C/D: VGPR v
// -> row m0+v (lanes 0-15) / m0+v+8 (lanes 16-31), col = n0 + (lane&15).
// ---------------------------------------------------------------------------
__global__ void __launch_bounds__(256) k_gemm_xlxr(const float* __restrict__ h,
                                                   const float* __restrict__ Wl,
                                                   const float* __restrict__ Wr,
                                                   float* __restrict__ xl,
                                                   float* __restrict__ xr,
                                                   int nNodes) {
  const int wave = blockIdx.x * (blockDim.x >> 5) + (threadIdx.x >> 5);
  const int lane = threadIdx.x & 31;
  const int m0 = wave * 16;
  if (m0 >= nNodes) return;  // wave-uniform: EXEC stays all-ones for WMMA

  const int laneM = lane & 15;        // row / col within tile
  const int laneH = (lane >> 4) << 1; // 0 for lanes 0-15, 2 for lanes 16-31

  v8f accL[8], accR[8];
  const v8f zero = {};
#pragma unroll
  for (int t = 0; t < 8; ++t) { accL[t] = zero; accR[t] = zero; }

  for (int k = 0; k < D; k += 4) {
    // A fragment: rows m0..m0+15, K = k..k+3
    const float* ap = h + (size_t)(m0 + laneM) * D + (k + laneH);
    v2f a;
    a.x = ap[0];
    a.y = ap[1];

#pragma unroll
    for (int t = 0; t < 8; ++t) {
      const int n0 = t * 16;
      const float* blp = Wl + (size_t)(k + laneH) * D + n0 + laneM;
      const float* brp = Wr + (size_t)(k + laneH) * D + n0 + laneM;
      v2f bl, br;
      bl.x = blp[0];
      bl.y = blp[D];
      br.x = brp[0];
      br.y = brp[D];
      accL[t] = __builtin_amdgcn_wmma_f32_16x16x4_f32(
          false, a, false, bl, (short)0, accL[t], false, false);
      accR[t] = __builtin_amdgcn_wmma_f32_16x16x4_f32(
          false, a, false, br, (short)0, accR[t], false, false);
    }
  }

  // Store C/D tiles
  const int rBase = m0 + ((lane >> 4) << 3);  // +0 or +8
#pragma unroll
  for (int t = 0; t < 8; ++t) {
    const int col = t * 16 + laneM;
#pragma unroll
    for (int v = 0; v < 8; ++v) {
      xl[(size_t)(rBase + v) * D + col] = accL[t][v];
      xr[(size_t)(rBase + v) * D + col] = accR[t][v];
    }
  }
}

// ---------------------------------------------------------------------------
// CAS-based float atomic max (monotone, safe for -inf init).
// ---------------------------------------------------------------------------
__device__ inline void atomicMaxF(float* addr, float val) {
  unsigned int* ua = (unsigned int*)addr;
  unsigned int old = *ua;
  while (__uint_as_float(old) < val) {
    unsigned int assumed = old;
    old = atomicCAS(ua, assumed, __float_as_uint(val));
    if (old == assumed) break;
  }
}

// ---------------------------------------------------------------------------
// Edge pass A: one wave per edge.  m = leakyrelu(xl[src]+xr[dst]+ea*We);
// logit = dot(m, att); store logit; atomic segment-max into mx[dst].
// ---------------------------------------------------------------------------
__global__ void __launch_bounds__(256) k_edge_logits(
    const float* __restrict__ xl, const float* __restrict__ xr,
    const float* __restrict__ eattr, const float* __restrict__ We,
    const float* __restrict__ att, const int* __restrict__ src,
    const int* __restrict__ dst, float* __restrict__ logits,
    float* __restrict__ mx, int nE) {
  const int e = blockIdx.x * (blockDim.x >> 5) + (threadIdx.x >> 5);
  if (e >= nE) return;
  const int lane = threadIdx.x & 31;
  const int s = src[e];
  const int d = dst[e];
  const float ea = eattr[e];
  const int c = lane * 4;

  const float4 a = *(const float4*)(xl + (size_t)s * D + c);
  const float4 b = *(const float4*)(xr + (size_t)d * D + c);
  const float4 w = *(const float4*)(We + c);
  const float4 av = *(const float4*)(att + c);

  float p = 0.0f;
  float m;
  m = a.x + b.x + ea * w.x; m = m > 0.0f ? m : NEG_SLOPE * m; p += m * av.x;
  m = a.y + b.y + ea * w.y; m = m > 0.0f ? m : NEG_SLOPE * m; p += m * av.y;
  m = a.z + b.z + ea * w.z; m = m > 0.0f ? m : NEG_SLOPE * m; p += m * av.z;
  m = a.w + b.w + ea * w.w; m = m > 0.0f ? m : NEG_SLOPE * m; p += m * av.w;

#pragma unroll
  for (int off = 16; off > 0; off >>= 1) p += __shfl_xor(p, off, 32);

  if (lane == 0) {
    logits[e] = p;
    atomicMaxF(mx + d, p);
  }
}

// ---------------------------------------------------------------------------
// Edge pass B: ex = exp(logit - mx[dst]); denom[dst] += ex (atomic).
// logits[] is overwritten with ex in place.
// ---------------------------------------------------------------------------
__global__ void __launch_bounds__(256) k_edge_exp(float* __restrict__ logits,
                                                  const float* __restrict__ mx,
                                                  const int* __restrict__ dst,
                                                  float* __restrict__ denom,
                                                  int nE) {
  const int e = blockIdx.x * blockDim.x + threadIdx.x;
  if (e >= nE) return;
  const int d = dst[e];
  const float ex = __expf(logits[e] - mx[d]);
  logits[e] = ex;
  atomicAdd(denom + d, ex);
}

// ---------------------------------------------------------------------------
// Edge pass C: one wave per edge.  alpha = ex / max(denom,1e-16);
// acc[dst] += alpha * xl[src]  (128 float atomics per edge, 4 per lane).
// ---------------------------------------------------------------------------
__global__ void __launch_bounds__(256) k_edge_scatter(
    const float* __restrict__ xl, const float* __restrict__ ex,
    const float* __restrict__ denom, const int* __restrict__ src,
    const int* __restrict__ dst, float* __restrict__ acc, int nE) {
  const int e = blockIdx.x * (blockDim.x >> 5) + (threadIdx.x >> 5);
  if (e >= nE) return;
  const int lane = threadIdx.x & 31;
  const int s = src[e];
  const int d = dst[e];
  const float alpha = ex[e] / fmaxf(denom[d], 1e-16f);
  const int c = lane * 4;
  const float4 a = *(const float4*)(xl + (size_t)s * D + c);
  float* ap = acc + (size_t)d * D + c;
  atomicAdd(ap + 0, alpha * a.x);
  atomicAdd(ap + 1, alpha * a.y);
  atomicAdd(ap + 2, alpha * a.z);
  atomicAdd(ap + 3, alpha * a.w);
}

// ---------------------------------------------------------------------------
// Finalize: hout = relu(acc + bias[col]).
// ---------------------------------------------------------------------------
__global__ void __launch_bounds__(256) k_finalize(const float* __restrict__ acc,
                                                  const float* __restrict__ bias,
                                                  float* __restrict__ hout,
                                                  int total) {
  const int i = blockIdx.x * blockDim.x + threadIdx.x;
  if (i >= total) return;
  const float v = acc[i] + bias[i & (D - 1)];
  hout[i] = v > 0.0f ? v : 0.0f;
}

// ---------------------------------------------------------------------------
// Host orchestration.
// Inputs (setup_inputs order): x, edge_index, edge_attr, Wl, Wr, We, att, bias
// ---------------------------------------------------------------------------
extern "C" void kernel_launch(void* const* d_in, const int* in_sizes, int n_in,
                              void* d_out, int out_size, void* d_ws, size_t ws_size,
                              hipStream_t stream) {
  const float* x     = (const float*)d_in[0];
  const int*   ei    = (const int*)d_in[1];
  const float* eattr = (const float*)d_in[2];
  const float* Wl    = (const float*)d_in[3];
  const float* Wr    = (const float*)d_in[4];
  const float* We    = (const float*)d_in[5];
  const float* att   = (const float*)d_in[6];
  const float* bias  = (const float*)d_in[7];
  float* out = (float*)d_out;

  const int N = in_sizes[0] / D;   // 50000
  const int E = in_sizes[2];       // 600000 (EDGE_DIM == 1)
  const int* srcIdx = ei;
  const int* dstIdx = ei + E;

  // Workspace carve-up
  float* ws = (float*)d_ws;
  float* xl     = ws; ws += (size_t)N * D;
  float* xr     = ws; ws += (size_t)N * D;
  float* acc    = ws; ws += (size_t)N * D;
  float* h1     = ws; ws += (size_t)N * D;
  float* logits = ws; ws += (size_t)E;
  float* mx     = ws; ws += (size_t)N;
  float* denom  = ws; ws += (size_t)N;

  const int total = N * D;
  const int blkElems  = (total + 255) / 256;
  const int blkGemm   = ((N + 15) / 16 + 7) / 8;  // 8 waves per 256-thread block
  const int blkEdgeW  = (E + 7) / 8;              // one wave per edge
  const int blkEdgeT  = (E + 255) / 256;          // one thread per edge

  for (int l = 0; l < 2; ++l) {
    const float* hin = (l == 0) ? x : h1;
    float* hout      = (l == 0) ? h1 : out;
    const float* Wl_l   = Wl + (size_t)l * D * D;
    const float* Wr_l   = Wr + (size_t)l * D * D;
    const float* We_l   = We + (size_t)l * D;     // EDGE_DIM=1 -> [1,D] row
    const float* att_l  = att + (size_t)l * D;
    const float* bias_l = bias + (size_t)l * D;

    k_init<<<blkElems, 256, 0, stream>>>(acc, mx, denom, N, total);
    k_gemm_xlxr<<<blkGemm, 256, 0, stream>>>(hin, Wl_l, Wr_l, xl, xr, N);
    k_edge_logits<<<blkEdgeW, 256, 0, stream>>>(xl, xr, eattr, We_l, att_l,
                                                srcIdx, dstIdx, logits, mx, E);
    k_edge_exp<<<blkEdgeT, 256, 0, stream>>>(logits, mx, dstIdx, denom, E);
    k_edge_scatter<<<blkEdgeW, 256, 0, stream>>>(xl, logits, denom, srcIdx,
                                                 dstIdx, acc, E);
    k_finalize<<<blkElems, 256, 0, stream>>>(acc, bias_l, hout, total);
  }
}